// PolymerGNN_66013647339945
// MI455X (gfx1250) — compile-verified
//
#include <hip/hip_runtime.h>
#include <hip/hip_bf16.h>
#include <math.h>

// ---------------------------------------------------------------------------
// PolymerGNN forward on gfx1250 (MI455X). fp32 end-to-end; GEMMs use
// V_WMMA_F32_16X16X4_F32 (exact fp32). 64-row blocks amortize B-fragment
// loads over 8 WMMAs. Edge scatter (HBM-bound) uses global_atomic_add_f32.
// ---------------------------------------------------------------------------

typedef __attribute__((ext_vector_type(2))) float v2f;
typedef __attribute__((ext_vector_type(8))) float v8f;

#define HID 128
#define HEADS 4
#define DH 32
#define NLAY 4
#define NEG_SLOPE 0.2f
#define LN_EPS 1e-5f

static __device__ inline v8f wmma_f32_k4(v2f a, v2f b, v8f c) {
    // D = A(16x4) * B(4x16) + C(16x16), fp32
    return __builtin_amdgcn_wmma_f32_16x16x4_f32(
        /*neg_a=*/false, a, /*neg_b=*/false, b,
        /*c_mod=*/(short)0, c, /*reuse_a=*/false, /*reuse_b=*/false);
}

static __device__ inline float waveReduceSum(float v) {
    #pragma unroll
    for (int o = 16; o; o >>= 1) v += __shfl_xor(v, o, 32);
    return v;
}

static __device__ inline float atomicMaxF32(float* addr, float val) {
    // sign-aware ordered-int trick; valid when buffer initialized to -inf
    if (val >= 0.0f)
        return __int_as_float(atomicMax((int*)addr, __float_as_int(val)));
    else
        return __uint_as_float(atomicMin((unsigned int*)addr, __float_as_uint(val)));
}

static __device__ inline float softplusf(float x) {
    return (x > 20.0f) ? x : log1pf(__expf(x));
}

// ------------------------------- fill --------------------------------------
__global__ void fill_kernel(float* __restrict__ p, float v, long long n) {
    long long i = (long long)blockIdx.x * blockDim.x + threadIdx.x;
    if (i < n) p[i] = v;
}

// ------------------------- encoder first layer (K=21) ----------------------
__global__ __launch_bounds__(128)
void enc1_kernel(const float* __restrict__ x, const float* __restrict__ W1,
                 const float* __restrict__ b1, float* __restrict__ t, int Nn) {
    int n = blockIdx.x;
    if (n >= Nn) return;
    int c = threadIdx.x;              // 0..127
    const float* xr = x + (long long)n * 21;
    float acc = b1[c];
    #pragma unroll
    for (int k = 0; k < 21; ++k) acc = fmaf(xr[k], W1[k * HID + c], acc);
    t[(long long)n * HID + c] = fmaxf(acc, 0.0f);
}

// ---------------------- WMMA fp32 GEMM: out = A[M,128] @ W[128,128] --------
// block = 128 threads (4 waves); 64-row block. wave w -> column tiles
// [32w, 32w+32); 4 row tiles per wave -> each B fragment feeds 4 WMMAs.
__global__ __launch_bounds__(128)
void gemm128_wmma(const float* __restrict__ A, const float* __restrict__ W,
                  const float* __restrict__ bias, float* __restrict__ out,
                  int M, int doRelu) {
    __shared__ float As[64][132];     // stride 132 -> conflict-free column reads
    int row0 = blockIdx.x * 64;
    int tid  = threadIdx.x;
    // cooperative float4 load of the 64x128 A tile (global_load_b128)
    for (int i = tid; i < 64 * 32; i += 128) {
        int r = i >> 5, c4 = i & 31;
        int row = row0 + r;
        if (row >= M) row = M - 1;    // clamp (stores are guarded)
        float4 v = ((const float4*)(A + (long long)row * HID))[c4];
        As[r][c4 * 4 + 0] = v.x;
        As[r][c4 * 4 + 1] = v.y;
        As[r][c4 * 4 + 2] = v.z;
        As[r][c4 * 4 + 3] = v.w;
    }
    __syncthreads();

    int wave = tid >> 5;
    int lane = tid & 31;
    int half = lane >> 4;             // 0: K0,K1 / 1: K2,K3
    int l16  = lane & 15;
    int nb0  = wave * 32;             // two 16-wide column tiles: nb0, nb0+16

    v8f acc[4][2] = {};
    #pragma unroll 2
    for (int s = 0; s < 32; ++s) {
        int k0 = s * 4;
        const float* Wr0 = W + (long long)(k0 + 2 * half) * HID;
        const float* Wr1 = Wr0 + HID;
        v2f b0 = { Wr0[nb0 + l16],      Wr1[nb0 + l16] };
        v2f b1 = { Wr0[nb0 + 16 + l16], Wr1[nb0 + 16 + l16] };
        #pragma unroll
        for (int rt = 0; rt < 4; ++rt) {
            v2f a = { As[rt * 16 + l16][k0 + 2 * half],
                      As[rt * 16 + l16][k0 + 2 * half + 1] };
            acc[rt][0] = wmma_f32_k4(a, b0, acc[rt][0]);
            acc[rt][1] = wmma_f32_k4(a, b1, acc[rt][1]);
        }
    }

    #pragma unroll
    for (int rt = 0; rt < 4; ++rt) {
        #pragma unroll
        for (int v = 0; v < 8; ++v) {
            int row = row0 + rt * 16 + v + half * 8;
            if (row >= M) continue;
            int col = nb0 + l16;
            float x0 = acc[rt][0][v], x1 = acc[rt][1][v];
            if (bias) { x0 += bias[col]; x1 += bias[col + 16]; }
            if (doRelu) { x0 = fmaxf(x0, 0.0f); x1 = fmaxf(x1, 0.0f); }
            out[(long long)row * HID + col]      = x0;
            out[(long long)row * HID + col + 16] = x1;
        }
    }
}

// ---------- per-node attention coefficients + per-layer buffer init --------
// one wave per node: al_src[n,h] = <xp[n,h*32:..], att_src[h]>, same for dst;
// also initializes m=-inf, denom=0, agg=0 for this node (fused fills).
__global__ __launch_bounds__(256)
void attn_init_kernel(const float* __restrict__ xp,
                      const float* __restrict__ a_src,
                      const float* __restrict__ a_dst,
                      float* __restrict__ al_src, float* __restrict__ al_dst,
                      float* __restrict__ m, float* __restrict__ denom,
                      float* __restrict__ agg, int Nn) {
    int w    = (blockIdx.x * blockDim.x + threadIdx.x) >> 5;
    int lane = threadIdx.x & 31;
    if (w >= Nn) return;
    const float* row = xp + (long long)w * HID;
    #pragma unroll
    for (int j = 0; j < HEADS; ++j) {
        float v = row[j * DH + lane];
        float s = waveReduceSum(v * a_src[j * DH + lane]);
        float d = waveReduceSum(v * a_dst[j * DH + lane]);
        if (lane == 0) { al_src[w * HEADS + j] = s; al_dst[w * HEADS + j] = d; }
        agg[(long long)w * HID + j * DH + lane] = 0.0f;
    }
    if (lane < HEADS) {
        m[w * HEADS + lane]     = -INFINITY;
        denom[w * HEADS + lane] = 0.0f;
    }
}

// ----------------------------- edge pass 1: max ----------------------------
__global__ __launch_bounds__(256)
void edge_max_kernel(const int* __restrict__ ei,
                     const float* __restrict__ al_src,
                     const float* __restrict__ al_dst,
                     float* __restrict__ m, int E, int Nn) {
    int e = blockIdx.x * blockDim.x + threadIdx.x;
    int ET = E + Nn;
    if (e >= ET) return;
    int s = (e < E) ? ei[e]     : (e - E);
    int d = (e < E) ? ei[E + e] : (e - E);
    #pragma unroll
    for (int j = 0; j < HEADS; ++j) {
        float v = al_src[s * HEADS + j] + al_dst[d * HEADS + j];
        v = (v > 0.0f) ? v : NEG_SLOPE * v;
        atomicMaxF32(&m[d * HEADS + j], v);
    }
}

// --------------------------- edge pass 2: denom ----------------------------
__global__ __launch_bounds__(256)
void edge_denom_kernel(const int* __restrict__ ei,
                       const float* __restrict__ al_src,
                       const float* __restrict__ al_dst,
                       const float* __restrict__ m,
                       float* __restrict__ denom, int E, int Nn) {
    int e = blockIdx.x * blockDim.x + threadIdx.x;
    int ET = E + Nn;
    if (e >= ET) return;
    int s = (e < E) ? ei[e]     : (e - E);
    int d = (e < E) ? ei[E + e] : (e - E);
    #pragma unroll
    for (int j = 0; j < HEADS; ++j) {
        float v = al_src[s * HEADS + j] + al_dst[d * HEADS + j];
        v = (v > 0.0f) ? v : NEG_SLOPE * v;
        atomicAdd(&denom[d * HEADS + j], __expf(v - m[d * HEADS + j]));
    }
}

// ----------------- edge pass 3: weighted message scatter (hot) -------------
// one wave per edge; lane covers channels {j*32+lane}
__global__ __launch_bounds__(256)
void edge_msg_kernel(const int* __restrict__ ei,
                     const float* __restrict__ al_src,
                     const float* __restrict__ al_dst,
                     const float* __restrict__ m,
                     const float* __restrict__ denom,
                     const float* __restrict__ xp,
                     float* __restrict__ agg, int E, int Nn) {
    int w    = (blockIdx.x * blockDim.x + threadIdx.x) >> 5;
    int lane = threadIdx.x & 31;
    int ET = E + Nn;
    if (w >= ET) return;
    int s = (w < E) ? ei[w]     : (w - E);
    int d = (w < E) ? ei[E + w] : (w - E);
    float alpha[HEADS];
    #pragma unroll
    for (int j = 0; j < HEADS; ++j) {
        float v = al_src[s * HEADS + j] + al_dst[d * HEADS + j];
        v = (v > 0.0f) ? v : NEG_SLOPE * v;
        alpha[j] = __expf(v - m[d * HEADS + j]) / (denom[d * HEADS + j] + 1e-16f);
    }
    const float* xr = xp + (long long)s * HID;
    float* ar = agg + (long long)d * HID;
    #pragma unroll
    for (int j = 0; j < HEADS; ++j)
        atomicAdd(&ar[j * DH + lane], xr[j * DH + lane] * alpha[j]);
}

// ------------- residual + bias + LayerNorm + ReLU (wave per node) ----------
__global__ __launch_bounds__(256)
void node_update_kernel(const float* __restrict__ agg,
                        const float* __restrict__ gb,
                        const float* __restrict__ lnw,
                        const float* __restrict__ lnb,
                        float* __restrict__ h, int Nn) {
    int n    = (blockIdx.x * blockDim.x + threadIdx.x) >> 5;
    int lane = threadIdx.x & 31;
    if (n >= Nn) return;
    float v[4], sum = 0.0f;
    #pragma unroll
    for (int j = 0; j < 4; ++j) {
        int c = j * DH + lane;
        v[j] = agg[(long long)n * HID + c] + gb[c] + h[(long long)n * HID + c];
        sum += v[j];
    }
    float mu = waveReduceSum(sum) * (1.0f / HID);
    float sq = 0.0f;
    #pragma unroll
    for (int j = 0; j < 4; ++j) { float dlt = v[j] - mu; sq += dlt * dlt; }
    float var = waveReduceSum(sq) * (1.0f / HID);
    float rs = rsqrtf(var + LN_EPS);
    #pragma unroll
    for (int j = 0; j < 4; ++j) {
        int c = j * DH + lane;
        float o = (v[j] - mu) * rs * lnw[c] + lnb[c];
        h[(long long)n * HID + c] = fmaxf(o, 0.0f);
    }
}

// ------------------------------ pooling ------------------------------------
__global__ __launch_bounds__(256)
void pool_add_kernel(const float* __restrict__ h, const int* __restrict__ batch,
                     float* __restrict__ h_graph, float* __restrict__ counts,
                     int Nn) {
    int n    = (blockIdx.x * blockDim.x + threadIdx.x) >> 5;
    int lane = threadIdx.x & 31;
    if (n >= Nn) return;
    int g = batch[n];
    #pragma unroll
    for (int j = 0; j < 4; ++j) {
        int c = j * DH + lane;
        atomicAdd(&h_graph[(long long)g * HID + c], h[(long long)n * HID + c]);
    }
    if (lane == 0) atomicAdd(&counts[g], 1.0f);
}

// ----------------------- per-graph heads + physics -------------------------
__global__ __launch_bounds__(128)
void head_kernel(const float* __restrict__ h_graph, const float* __restrict__ counts,
                 const float* __restrict__ gfeat,
                 const float* __restrict__ globW, const float* __restrict__ globb,
                 const float* __restrict__ omW1, const float* __restrict__ omb1,
                 const float* __restrict__ omW2, const float* __restrict__ omb2,
                 const float* __restrict__ cvW1, const float* __restrict__ cvb1,
                 const float* __restrict__ cvW2, const float* __restrict__ cvb2,
                 const float* __restrict__ log_lambda, const float* __restrict__ log_alpha,
                 const float* __restrict__ log_omega_max,
                 float* __restrict__ out, int G, int GFdim) {
    int g = blockIdx.x;
    if (g >= G) return;
    int t = threadIdx.x;                        // 0..127
    __shared__ float hc[2 * HID];
    __shared__ float red[HID];
    __shared__ float om_s;

    float cnt = fmaxf(counts[g], 1.0f);
    hc[t] = h_graph[(long long)g * HID + t] / cnt;
    float acc = globb[t];
    for (int k = 0; k < GFdim; ++k) acc = fmaf(gfeat[g * GFdim + k], globW[k * HID + t], acc);
    hc[HID + t] = acc;
    __syncthreads();

    // omega head: relu(hc @ W1 + b1) @ W2 + b2 -> softplus
    float a1 = omb1[t];
    for (int k = 0; k < 2 * HID; ++k) a1 = fmaf(hc[k], omW1[k * HID + t], a1);
    a1 = fmaxf(a1, 0.0f);
    red[t] = a1 * omW2[t];
    __syncthreads();
    for (int sft = 64; sft; sft >>= 1) { if (t < sft) red[t] += red[t + sft]; __syncthreads(); }
    if (t == 0) om_s = softplusf(red[0] + omb2[0]);
    __syncthreads();

    // cv head
    float c1 = cvb1[t];
    for (int k = 0; k < 2 * HID; ++k) c1 = fmaf(hc[k], cvW1[k * HID + t], c1);
    c1 = fmaxf(c1, 0.0f);
    red[t] = c1 * cvW2[t];
    __syncthreads();
    for (int sft = 64; sft; sft >>= 1) { if (t < sft) red[t] += red[t + sft]; __syncthreads(); }
    if (t == 0) {
        float cv_direct = softplusf(red[0] + cvb2[0]);
        float alpha     = __expf(log_alpha[0]);
        float omega_max = __expf(log_omega_max[0]);
        float omega_eff = fmaxf(fminf(alpha * om_s, omega_max), 2.0f);
        float lmbda     = __expf(log_lambda[0]);
        float causality = __expf(-lmbda * __logf(omega_eff));   // omega_eff^-lambda, omega_eff>=2
        float cv_phys   = 30.0f * (1.0f - causality);
        out[g] = 0.5f * cv_phys + 0.5f * cv_direct;
    }
}

// ---------------------------------------------------------------------------
extern "C" void kernel_launch(void* const* d_in, const int* in_sizes, int n_in,
                              void* d_out, int out_size, void* d_ws, size_t ws_size,
                              hipStream_t stream) {
    const float* x      = (const float*)d_in[0];
    const int*   ei     = (const int*)d_in[1];
    const int*   batch  = (const int*)d_in[2];
    const float* gfeat  = (const float*)d_in[3];
    const float* enc_W1 = (const float*)d_in[4];
    const float* enc_b1 = (const float*)d_in[5];
    const float* enc_W2 = (const float*)d_in[6];
    const float* enc_b2 = (const float*)d_in[7];
    const float* gat_W  = (const float*)d_in[8];    // [4,128,128]
    const float* a_src  = (const float*)d_in[9];    // [4,4,32]
    const float* a_dst  = (const float*)d_in[10];
    const float* gat_b  = (const float*)d_in[11];   // [4,128]
    const float* ln_w   = (const float*)d_in[12];
    const float* ln_b   = (const float*)d_in[13];
    const float* glob_W = (const float*)d_in[14];
    const float* glob_b = (const float*)d_in[15];
    const float* om_W1  = (const float*)d_in[16];
    const float* om_b1  = (const float*)d_in[17];
    const float* om_W2  = (const float*)d_in[18];
    const float* om_b2  = (const float*)d_in[19];
    const float* cv_W1  = (const float*)d_in[20];
    const float* cv_b1  = (const float*)d_in[21];
    const float* cv_W2  = (const float*)d_in[22];
    const float* cv_b2  = (const float*)d_in[23];
    const float* log_lambda    = (const float*)d_in[24];
    const float* log_alpha     = (const float*)d_in[25];
    const float* log_omega_max = (const float*)d_in[26];
    float* out = (float*)d_out;

    const int N  = in_sizes[2];           // 100000
    const int E  = in_sizes[1] / 2;       // 1600000
    const int GFdim = 4;
    const int G  = in_sizes[3] / GFdim;   // 512
    const int ET = E + N;                 // edges incl. self loops

    // workspace layout (floats)
    float* ws = (float*)d_ws;
    size_t off = 0;
    float* h       = ws + off; off += (size_t)N * HID;
    float* xp      = ws + off; off += (size_t)N * HID;   // also enc hidden
    float* agg     = ws + off; off += (size_t)N * HID;
    float* al_src  = ws + off; off += (size_t)N * HEADS;
    float* al_dst  = ws + off; off += (size_t)N * HEADS;
    float* mbuf    = ws + off; off += (size_t)N * HEADS;
    float* denom   = ws + off; off += (size_t)N * HEADS;
    float* h_graph = ws + off; off += (size_t)G * HID;
    float* counts  = ws + off; off += (size_t)G;
    (void)ws_size;

    const int nodeWaveBlocks = (N + 7) / 8;               // 8 waves / 256-thread block
    const int edgeWaveBlocks = (ET + 7) / 8;
    const int edgeThrBlocks  = (ET + 255) / 256;
    const int mTiles         = (N + 63) / 64;

    // -------- encoder --------
    enc1_kernel<<<N, 128, 0, stream>>>(x, enc_W1, enc_b1, xp, N);
    gemm128_wmma<<<mTiles, 128, 0, stream>>>(xp, enc_W2, enc_b2, h, N, 0);

    // -------- GAT layers --------
    for (int l = 0; l < NLAY; ++l) {
        gemm128_wmma<<<mTiles, 128, 0, stream>>>(h, gat_W + (size_t)l * HID * HID,
                                                 nullptr, xp, N, 0);
        attn_init_kernel<<<nodeWaveBlocks, 256, 0, stream>>>(
            xp, a_src + l * HEADS * DH, a_dst + l * HEADS * DH,
            al_src, al_dst, mbuf, denom, agg, N);

        edge_max_kernel  <<<edgeThrBlocks, 256, 0, stream>>>(ei, al_src, al_dst, mbuf, E, N);
        edge_denom_kernel<<<edgeThrBlocks, 256, 0, stream>>>(ei, al_src, al_dst, mbuf, denom, E, N);
        edge_msg_kernel  <<<edgeWaveBlocks, 256, 0, stream>>>(ei, al_src, al_dst, mbuf, denom,
                                                              xp, agg, E, N);
        node_update_kernel<<<nodeWaveBlocks, 256, 0, stream>>>(
            agg, gat_b + l * HID, ln_w + l * HID, ln_b + l * HID, h, N);
    }

    // -------- pooling + heads --------
    long long gh = (long long)G * HID;
    fill_kernel<<<(unsigned)((gh + 255) / 256), 256, 0, stream>>>(h_graph, 0.0f, gh);
    fill_kernel<<<(unsigned)((G + 255) / 256), 256, 0, stream>>>(counts, 0.0f, G);
    pool_add_kernel<<<nodeWaveBlocks, 256, 0, stream>>>(h, batch, h_graph, counts, N);

    head_kernel<<<G, 128, 0, stream>>>(h_graph, counts, gfeat, glob_W, glob_b,
                                       om_W1, om_b1, om_W2, om_b2,
                                       cv_W1, cv_b1, cv_W2, cv_b2,
                                       log_lambda, log_alpha, log_omega_max,
                                       out, G, GFdim);
}